// MPLayer_42460046688999
// MI455X (gfx1250) — compile-verified
//
#include <hip/hip_runtime.h>
#include <stdint.h>

// ---------------------------------------------------------------------------
// MPLayer: for each edge (u,v) of relation r:
//   y[b,v] += p[r]*x[b,u];  y[b,u] += p[r]*x[b,v]   (b = 0..3)
// Strategy: repack x -> xt[N][4], zero yt[N][4]; stream edges HBM->LDS with
// non-temporal async-to-LDS DMA (double buffered, ASYNCcnt); gather float4
// from xt (L2-hot); scatter with native f32 L2 atomics (immediate offsets);
// unpack yt -> out[B][N].
// ---------------------------------------------------------------------------

#define TILE_EDGES      128           // edges per wave-tile (4 b128 per lane)
#define LOADS_PER_FILL  4
#define BLOCK           256
#define WAVES_PER_BLOCK (BLOCK / 32)
#define EDGE_BLOCKS     1024

// Low 32 bits of a generic pointer to __shared__ == LDS byte offset (DS addr).
__device__ __forceinline__ unsigned lds_off(const void* p) {
  return (unsigned)(uintptr_t)p;
}

// CDNA5 async global->LDS copy, 16B/lane, ASYNCcnt-tracked, non-temporal
// (edge stream is single-pass; keep L2 for the x/y working set).
__device__ __forceinline__ void async_copy_b128_nt(unsigned lds, const void* g) {
  asm volatile("global_load_async_to_lds_b128 %0, %1, off th:TH_LOAD_NT"
               :: "v"(lds), "v"((unsigned long long)(uintptr_t)g)
               : "memory");
}
__device__ __forceinline__ void wait_async_le4() {
  asm volatile("s_wait_asynccnt 4" ::: "memory");
}
__device__ __forceinline__ void wait_async_0() {
  asm volatile("s_wait_asynccnt 0" ::: "memory");
}

// Native device-scope f32 atomic add with immediate byte offset (one 64-bit
// address pair feeds 4 consecutive-float atomics).
template <int OFF>
__device__ __forceinline__ void gatomic_add_f32(const float* base, float v) {
  asm volatile("global_atomic_add_f32 %0, %1, off offset:%2 scope:SCOPE_DEV"
               :: "v"((unsigned long long)(uintptr_t)base), "v"(v), "i"(OFF)
               : "memory");
}

// ------------------------- layout kernels ----------------------------------

__global__ void mp_pack_x(const float* __restrict__ x, float4* __restrict__ xt,
                          float4* __restrict__ yt, int N) {
  int i = blockIdx.x * blockDim.x + threadIdx.x;
  if (i < N) {
    float4 t;
    t.x = __builtin_nontemporal_load(x + i);               // x read once
    t.y = __builtin_nontemporal_load(x + i + (size_t)N);
    t.z = __builtin_nontemporal_load(x + i + 2 * (size_t)N);
    t.w = __builtin_nontemporal_load(x + i + 3 * (size_t)N);
    xt[i] = t;                                             // reread: keep RT
    yt[i] = make_float4(0.f, 0.f, 0.f, 0.f);
  }
}

__global__ void mp_unpack_y(const float4* __restrict__ yt,
                            float* __restrict__ out, int N) {
  int i = blockIdx.x * blockDim.x + threadIdx.x;
  if (i < N) {
    float4 v = yt[i];                                      // last use
    __builtin_nontemporal_store(v.x, out + i);             // out never reread
    __builtin_nontemporal_store(v.y, out + i + (size_t)N);
    __builtin_nontemporal_store(v.z, out + i + 2 * (size_t)N);
    __builtin_nontemporal_store(v.w, out + i + 3 * (size_t)N);
  }
}

// ------------------------- edge scatter kernel -----------------------------

template <bool TR>
__device__ __forceinline__ void do_edge(unsigned u, unsigned v, float w,
                                        const float* __restrict__ xs,
                                        float* __restrict__ yd, int N) {
  if (TR) {
    // xs = xt[N][4], yd = yt[N][4]: 2 line-local loads + 2 line-local scatters
    const float4* xt = (const float4*)xs;
    float4 xu = xt[u];
    float4 xv = xt[v];
    const float* yv = yd + 4ull * v;
    const float* yu = yd + 4ull * u;
    gatomic_add_f32<0>(yv, w * xu.x);
    gatomic_add_f32<4>(yv, w * xu.y);
    gatomic_add_f32<8>(yv, w * xu.z);
    gatomic_add_f32<12>(yv, w * xu.w);
    gatomic_add_f32<0>(yu, w * xv.x);
    gatomic_add_f32<4>(yu, w * xv.y);
    gatomic_add_f32<8>(yu, w * xv.z);
    gatomic_add_f32<12>(yu, w * xv.w);
  } else {
#pragma unroll
    for (int b = 0; b < 4; ++b) {
      float xu = xs[(size_t)b * N + u];
      float xv = xs[(size_t)b * N + v];
      gatomic_add_f32<0>(yd + (size_t)b * N + v, w * xu);
      gatomic_add_f32<0>(yd + (size_t)b * N + u, w * xv);
    }
  }
}

template <bool TR>
__global__ void __launch_bounds__(BLOCK)
mp_edges(const ulonglong2* __restrict__ edges, const float* __restrict__ p,
         const float* __restrict__ xs, float* __restrict__ yd,
         unsigned totalEdges, unsigned Epr, int N) {
  __shared__ ulonglong2 ebuf[WAVES_PER_BLOCK * 2 * TILE_EDGES];

  const int wave = threadIdx.x >> 5;
  const int lane = threadIdx.x & 31;
  ulonglong2* wbuf = &ebuf[wave * 2 * TILE_EDGES];

  const unsigned tiles    = totalEdges / TILE_EDGES;
  const unsigned numWaves = gridDim.x * WAVES_PER_BLOCK;
  const unsigned gw       = blockIdx.x * WAVES_PER_BLOCK + wave;

  auto fill = [&](unsigned tile, int b) {
    const ulonglong2* src = edges + (size_t)tile * TILE_EDGES + lane;
    ulonglong2* dst = wbuf + b * TILE_EDGES + lane;
#pragma unroll
    for (int j = 0; j < LOADS_PER_FILL; ++j)
      async_copy_b128_nt(lds_off(dst + j * 32), src + j * 32);
  };

  int buf = 0;
  if (gw < tiles) fill(gw, buf);

  for (unsigned t = gw; t < tiles; t += numWaves) {
    const unsigned nt = t + numWaves;
    const bool more = nt < tiles;
    if (more) {
      fill(nt, buf ^ 1);
      wait_async_le4();   // current tile's 4 DMAs complete (in-order)
    } else {
      wait_async_0();
    }

    // Relation weight: one scalar div per tile; a 128-edge tile crosses at
    // most one relation boundary (E >> 128).
    const unsigned base = t * TILE_EDGES;
    const unsigned r0   = base / Epr;
    const unsigned rem0 = base - r0 * Epr;

#pragma unroll
    for (int j = 0; j < LOADS_PER_FILL; ++j) {
      const unsigned ii = (unsigned)(j * 32 + lane);
      ulonglong2 e = wbuf[buf * TILE_EDGES + ii];
      const unsigned u = (unsigned)e.x;
      const unsigned v = (unsigned)e.y;
      const float w = p[r0 + ((rem0 + ii) >= Epr ? 1u : 0u)];
      do_edge<TR>(u, v, w, xs, yd, N);
    }
    buf ^= 1;
  }

  // Tail edges (totalEdges % TILE_EDGES), handled directly from global.
  const unsigned tailStart = tiles * TILE_EDGES;
  const unsigned gtid   = blockIdx.x * blockDim.x + threadIdx.x;
  const unsigned stride = gridDim.x * blockDim.x;
  for (unsigned i = tailStart + gtid; i < totalEdges; i += stride) {
    ulonglong2 e = edges[i];
    do_edge<TR>((unsigned)e.x, (unsigned)e.y, p[i / Epr], xs, yd, N);
  }
}

// ------------------------------ launcher -----------------------------------

extern "C" void kernel_launch(void* const* d_in, const int* in_sizes, int n_in,
                              void* d_out, int out_size, void* d_ws,
                              size_t ws_size, hipStream_t stream) {
  const float*      p     = (const float*)d_in[0];
  const ulonglong2* edges = (const ulonglong2*)d_in[1];  // [R*E] pairs of i64
  const float*      x     = (const float*)d_in[2];       // [B, N]
  float*            out   = (float*)d_out;               // [B, N]

  const int      B          = 4;
  const int      R          = in_sizes[0];
  const int      N          = in_sizes[2] / B;
  const unsigned totalEdges = (unsigned)((size_t)in_sizes[1] / 2);
  const unsigned Epr        = totalEdges / (unsigned)R;

  const size_t need = 2ull * (size_t)N * B * sizeof(float);  // xt + yt
  const int nblk = (N + BLOCK - 1) / BLOCK;

  if (ws_size >= need) {
    float4* xt = (float4*)d_ws;
    float4* yt = xt + N;
    mp_pack_x<<<nblk, BLOCK, 0, stream>>>(x, xt, yt, N);
    mp_edges<true><<<EDGE_BLOCKS, BLOCK, 0, stream>>>(
        edges, p, (const float*)xt, (float*)yt, totalEdges, Epr, N);
    mp_unpack_y<<<nblk, BLOCK, 0, stream>>>(yt, out, N);
  } else {
    hipMemsetAsync(out, 0, (size_t)out_size * sizeof(float), stream);
    mp_edges<false><<<EDGE_BLOCKS, BLOCK, 0, stream>>>(
        edges, p, x, out, totalEdges, Epr, N);
  }
}